// DGMLayer_63084479644217
// MI455X (gfx1250) — compile-verified
//
#include <hip/hip_runtime.h>
#include <hip/hip_bf16.h>
#include <math.h>

typedef __attribute__((ext_vector_type(2))) float v2f;
typedef __attribute__((ext_vector_type(8))) float v8f;

#define BATCH  32
#define NNODES 1024
#define KSEL   16
#define CDIM   64
#define FDIM   128

// PCG-style hash -> uniform(0,1) -> Gumbel(0,1)
__device__ __forceinline__ float gumbel_noise(unsigned s) {
    s = s * 747796405u + 2891336453u;
    unsigned w = ((s >> ((s >> 28) + 4u)) ^ s) * 277803737u;
    w = (w >> 22) ^ w;
    float u = ((float)(w >> 8) + 0.5f) * (1.0f / 16777216.0f);
    return -__logf(-__logf(u));
}

// -------------------------------------------------------------------------
// Kernel 1: out = x @ W + b   ([32768,128] x [128,64]), f32 WMMA 16x16x4.
// One wave per 16x16 output tile; bias folded into accumulator init.
// -------------------------------------------------------------------------
__global__ void encoder_gemm(const float* __restrict__ x,
                             const float* __restrict__ W,
                             const float* __restrict__ bias,
                             float* __restrict__ out) {
    const int wave = threadIdx.x >> 5;
    const int lane = threadIdx.x & 31;
    const int half = lane >> 4;       // 0: lanes 0-15, 1: lanes 16-31
    const int nl   = lane & 15;       // row (A) / col (B,C) index within tile
    const int mbase = (blockIdx.x * 4 + wave) * 16;
    const int nbase = blockIdx.y * 16;

    // C init = bias (same for every row of the tile)
    float bn = bias[nbase + nl];
    v8f c;
#pragma unroll
    for (int r = 0; r < 8; ++r) c[r] = bn;

#pragma unroll
    for (int t = 0; t < FDIM / 4; ++t) {
        const int k = t * 4 + half * 2;
        // A[m, k..k+1]: contiguous float2 per lane
        v2f a = *(const v2f*)(x + (size_t)(mbase + nl) * FDIM + k);
        // B[k..k+1, n]: strided by CDIM
        v2f b;
        b.x = W[(size_t)k * CDIM + nbase + nl];
        b.y = W[(size_t)(k + 1) * CDIM + nbase + nl];
        c = __builtin_amdgcn_wmma_f32_16x16x4_f32(false, a, false, b,
                                                  (short)0, c, false, false);
    }

#pragma unroll
    for (int r = 0; r < 8; ++r) {
        const int m = r + half * 8;
        out[(size_t)(mbase + m) * CDIM + nbase + nl] = c[r];
    }
}

// -------------------------------------------------------------------------
// Kernel 2: sq[row] = sum_c out[row,c]^2   (one wave per row of 64)
// -------------------------------------------------------------------------
__global__ void row_sq(const float* __restrict__ out, float* __restrict__ sq) {
    const int wave = threadIdx.x >> 5;
    const int lane = threadIdx.x & 31;
    const int row  = blockIdx.x * 8 + wave;
    const float* p = out + (size_t)row * CDIM;
    float v0 = p[lane], v1 = p[lane + 32];
    float s = v0 * v0 + v1 * v1;
#pragma unroll
    for (int off = 16; off > 0; off >>= 1) s += __shfl_xor(s, off, 32);
    if (lane == 0) sq[row] = s;
}

// -------------------------------------------------------------------------
// Kernel 3: fused  d_e -> exp(-T*d) + gumbel -> top-16 per row.
// One workgroup (256 thr = 8 waves) per (graph, 16-row block).
//   Phase A: WMMA h_i . h_j^T, scores tile [16][1024] in LDS (never to HBM).
//   Phase B: 2 rows per wave; iterative argmax with register-resident row.
// -------------------------------------------------------------------------
__global__ void dist_topk(const float* __restrict__ h,
                          const float* __restrict__ sq,
                          const float* __restrict__ temp,
                          float* __restrict__ erow,
                          float* __restrict__ ecol) {
    extern __shared__ float smem[];
    float (*s_scores)[NNODES] = (float (*)[NNODES])smem;   // 16*1024 f32 = 64KB
    float* s_sqj = smem + 16 * NNODES;                      // 1024 f32
    float* s_sqi = s_sqj + NNODES;                          // 16 f32

    const int g     = blockIdx.y;
    const int ibase = blockIdx.x * 16;
    const int tid   = threadIdx.x;
    const int wave  = tid >> 5;
    const int lane  = tid & 31;
    const int half  = lane >> 4;
    const int nl    = lane & 15;

    for (int t = tid; t < NNODES; t += 256) s_sqj[t] = sq[g * NNODES + t];
    if (tid < 16) s_sqi[tid] = sq[g * NNODES + ibase + tid];
    __syncthreads();

    const float T = temp[0];
    const float* hg = h + (size_t)g * NNODES * CDIM;

    // Hoist A fragments (rows ibase..ibase+15, K=64): 16 x float2 per lane
    v2f a[16];
#pragma unroll
    for (int t = 0; t < 16; ++t) {
        const int k = t * 4 + half * 2;
        a[t] = *(const v2f*)(hg + (size_t)(ibase + nl) * CDIM + k);
    }

    // Phase A: each wave covers 8 of the 64 j-tiles (no divergence; EXEC all-1)
    for (int jt = wave; jt < NNODES / 16; jt += 8) {
        const int jbase = jt * 16;
        v8f c = {};
#pragma unroll
        for (int t = 0; t < 16; ++t) {
            const int k = t * 4 + half * 2;
            v2f b = *(const v2f*)(hg + (size_t)(jbase + nl) * CDIM + k);
            c = __builtin_amdgcn_wmma_f32_16x16x4_f32(false, a[t], false, b,
                                                      (short)0, c, false, false);
        }
#pragma unroll
        for (int r = 0; r < 8; ++r) {
            const int m = r + half * 8;
            const int j = jbase + nl;
            const float d = s_sqi[m] + s_sqj[j] - 2.0f * c[r];
            const unsigned key = (unsigned)((g * NNODES + ibase + m) * NNODES + j);
            s_scores[m][j] = __expf(-T * d) + gumbel_noise(key);
        }
    }
    __syncthreads();

    // Phase B: top-16 per row; 2 rows per wave
#pragma unroll
    for (int rr = 0; rr < 2; ++rr) {
        const int m    = wave * 2 + rr;
        const int grow = g * NNODES + ibase + m;

        float vals[32];
#pragma unroll
        for (int t = 0; t < 32; ++t) vals[t] = s_scores[m][lane + t * 32];

        for (int it = 0; it < KSEL; ++it) {
            // lane-local argmax over 32 register values
            float best = -INFINITY; int bt = 0;
#pragma unroll
            for (int t = 0; t < 32; ++t)
                if (vals[t] > best) { best = vals[t]; bt = t; }
            int bidx = bt * 32 + lane;
            // wave-wide argmax (ties -> lower index, matches top_k stability)
#pragma unroll
            for (int off = 16; off > 0; off >>= 1) {
                float ov = __shfl_xor(best, off, 32);
                int   oi = __shfl_xor(bidx, off, 32);
                if (ov > best || (ov == best && oi < bidx)) { best = ov; bidx = oi; }
            }
            // mask winner (pure v_cndmask; keeps vals in VGPRs, no spill)
            const int  wt = bidx >> 5;
            const bool wl = (lane == (bidx & 31));
#pragma unroll
            for (int t = 0; t < 32; ++t)
                vals[t] = (wl && t == wt) ? -INFINITY : vals[t];

            if (lane == 0) {
                const int e = grow * KSEL + it;
                erow[e] = (float)grow;
                ecol[e] = (float)(g * NNODES + bidx);
            }
        }
    }
}

// -------------------------------------------------------------------------
extern "C" void kernel_launch(void* const* d_in, const int* in_sizes, int n_in,
                              void* d_out, int out_size, void* d_ws, size_t ws_size,
                              hipStream_t stream) {
    (void)in_sizes; (void)n_in; (void)out_size; (void)ws_size;
    const float* x    = (const float*)d_in[0];   // [32768,128]
    const float* W    = (const float*)d_in[1];   // [128,64]
    const float* bias = (const float*)d_in[2];   // [64]
    const float* temp = (const float*)d_in[3];   // scalar

    float* out  = (float*)d_out;                            // [32768,64]
    float* erow = out + (size_t)BATCH * NNODES * CDIM;      // [B*N*K] rows
    float* ecol = erow + (size_t)BATCH * NNODES * KSEL;     // [B*N*K] cols
    float* sq   = (float*)d_ws;                             // [32768] scratch

    // 1) encoder GEMM: grid = (2048/4 m-tiles, 4 n-tiles), 4 waves/block
    encoder_gemm<<<dim3(512, 4), 128, 0, stream>>>(x, W, bias, out);

    // 2) row squared norms: one wave per row
    row_sq<<<dim3((BATCH * NNODES) / 8), 256, 0, stream>>>(out, sq);

    // 3) fused distance + gumbel + top-k: one block per (row-block, graph)
    const size_t smem = (size_t)(16 * NNODES + NNODES + 16) * sizeof(float);
    dist_topk<<<dim3(NNODES / 16, BATCH), 256, smem, stream>>>(out, sq, temp,
                                                               erow, ecol);
}